// _HashGridMLP_33706903339712
// MI455X (gfx1250) — compile-verified
//
#include <hip/hip_runtime.h>
#include <stdint.h>

// HashGrid gather for MI455X (gfx1250).
// Reference collapses to: out[p, 0..7] = embedding[hash(trunc(x[p]*128))].
// Pure bandwidth kernel: stage x via TDM DMA into LDS, gather 32B/point from
// the L2-resident 16MB table, stream results out with non-temporal stores.

#define HASHMAP_MASK  (524288u - 1u)   // 2^19 table
#define RES           128.0f
#define PTS_PER_BLOCK 1024
#define THREADS       256              // 8 wave32s per block

typedef float    v4f   __attribute__((ext_vector_type(4)));
typedef uint32_t u32x4 __attribute__((ext_vector_type(4)));
typedef int32_t  i32x8 __attribute__((ext_vector_type(8)));
typedef int32_t  i32x4 __attribute__((ext_vector_type(4)));

__global__ __launch_bounds__(THREADS) void hashgrid_gather_kernel(
    const float* __restrict__ x, const float* __restrict__ emb,
    float* __restrict__ out, int n)
{
    __shared__ __align__(16) float lds_x[3 * PTS_PER_BLOCK];  // 12 KB staging

    const int block0 = blockIdx.x * PTS_PER_BLOCK;
    int elems = 3 * (n - block0);                 // floats this block stages
    if (elems > 3 * PTS_PER_BLOCK) elems = 3 * PTS_PER_BLOCK;

    // ---- TDM: one bulk DMA of this block's x-slice (global -> LDS) --------
    // Issued once by wave 0 (EXEC is ignored by tensor ops; descriptor groups
    // are uniform -> SGPRs). D# layout per CDNA5 ISA ch. 8.
    if (threadIdx.x < 32 && elems > 0) {
        uint64_t gaddr = (uint64_t)(uintptr_t)(x + (size_t)block0 * 3);
        uint32_t laddr = (uint32_t)(uintptr_t)(void*)&lds_x[0]; // low 32b = LDS offset
        uint32_t ne    = (uint32_t)elems;                       // <= 3072, fits tile_dim0

        u32x4 g0;
        g0[0] = 1u;                                            // count=1, user descriptor
        g0[1] = laddr;                                         // lds_addr (bytes)
        g0[2] = (uint32_t)gaddr;                               // global_addr[31:0]
        g0[3] = ((uint32_t)(gaddr >> 32) & 0x01FFFFFFu)        // global_addr[56:32]
              | (2u << 30);                                    // type = 2 ("image")

        i32x8 g1;
        g1[0] = (int32_t)(2u << 16);                           // data_size=2 (4B), wg_mask=0
        g1[1] = (int32_t)((ne & 0xFFFFu) << 16);               // tensor_dim0[15:0] @ [31:16]
        g1[2] = (int32_t)(((ne >> 16) & 0xFFFFu) | (1u << 16));// tensor_dim0[31:16]; tensor_dim1=1
        g1[3] = (int32_t)((ne & 0xFFFFu) << 16);               // tile_dim0 = ne @ [31:16]
        g1[4] = 0;                                             // tile_dim1=0, tile_dim2=0 (unused)
        g1[5] = (int32_t)ne;                                   // tensor_dim0_stride[31:0]
        g1[6] = 0;                                             // stride hi / dim1_stride
        g1[7] = 0;

        i32x4 gz4 = {0, 0, 0, 0};                              // 1-D: groups 2/3 unused
        i32x8 gz8 = {0, 0, 0, 0, 0, 0, 0, 0};                  // extra group (clang-23 form)
        __builtin_amdgcn_tensor_load_to_lds(g0, g1, gz4, gz4, gz8, /*cpol=*/0);
        __builtin_amdgcn_s_wait_tensorcnt(0);                  // TENSORcnt == 0
    }
    __syncthreads();

    // ---- gather + stream out ----------------------------------------------
    const v4f* __restrict__ embv = (const v4f*)emb;   // 2 x v4f per table row
    v4f* __restrict__ outv       = (v4f*)out;

    #pragma unroll
    for (int k = 0; k < PTS_PER_BLOCK / THREADS; ++k) {
        const int pl = threadIdx.x + k * THREADS;     // local point
        const int p  = block0 + pl;                   // global point
        if (p < n) {
            // trunc(x*128) -> int32 -> uint32 (wrap semantics match jnp)
            uint32_t ix = (uint32_t)(int32_t)(lds_x[3 * pl + 0] * RES);
            uint32_t iy = (uint32_t)(int32_t)(lds_x[3 * pl + 1] * RES);
            uint32_t iz = (uint32_t)(int32_t)(lds_x[3 * pl + 2] * RES);
            // primes: 1, 2654435761, 805459861 ; xor-fold ; mask (2^19)
            uint32_t h   = ix ^ (iy * 2654435761u) ^ (iz * 805459861u);
            uint32_t idx = h & HASHMAP_MASK;

            // 32B gather from L2-resident table (RT policy: keep cached)
            v4f lo = embv[2 * (size_t)idx + 0];
            v4f hi = embv[2 * (size_t)idx + 1];

            // 32B streamed result: non-temporal so it doesn't evict the table
            __builtin_nontemporal_store(lo, &outv[2 * (size_t)p + 0]);
            __builtin_nontemporal_store(hi, &outv[2 * (size_t)p + 1]);
        }
    }
}

extern "C" void kernel_launch(void* const* d_in, const int* in_sizes, int n_in,
                              void* d_out, int out_size, void* d_ws, size_t ws_size,
                              hipStream_t stream) {
    const float* x   = (const float*)d_in[0];   // [N, 3] fp32
    const float* emb = (const float*)d_in[1];   // [524288, 8] fp32
    float* out       = (float*)d_out;           // [N, 8] fp32

    const int n      = in_sizes[0] / 3;
    const int blocks = (n + PTS_PER_BLOCK - 1) / PTS_PER_BLOCK;
    hashgrid_gather_kernel<<<blocks, THREADS, 0, stream>>>(x, emb, out, n);
}